// Model_70995809403513
// MI455X (gfx1250) — compile-verified
//
#include <hip/hip_runtime.h>

// ---------------- CDNA5 WMMA f32 16x16x4 ----------------
typedef __attribute__((ext_vector_type(2))) float v2f;
typedef __attribute__((ext_vector_type(8))) float v8f;
typedef __attribute__((ext_vector_type(4))) float f4;

#define WMMA4(a, b, c) \
  __builtin_amdgcn_wmma_f32_16x16x4_f32(false, (a), false, (b), (short)0, (c), false, false)

#define DD   768
#define HH   12
#define HD   64
#define BB   32
#define SEQ  128
#define SS   127

// ws layout (floats)
#define VWS_OFF   0            // 32*12*128*64 = 3145728
#define WEFF_OFF  3145728      // 2*12*768     = 18432
#define LQ_OFF    3164160      // 32*12*128    = 49152  (lk at +49152)

// ---------------------------------------------------------------
// Kernel A: fold attention vector into W:  weff[qk][h][c] = sum_d Wa[qk*64+d] * W[h*64+d][c]
__global__ __launch_bounds__(256) void weff_kernel(const float* __restrict__ Wq,
                                                   const float* __restrict__ Wk,
                                                   const float* __restrict__ Wa,
                                                   float* __restrict__ weff) {
  int idx = blockIdx.x * 256 + threadIdx.x;         // 0 .. 2*12*768-1
  if (idx >= 2 * HH * DD) return;
  int qk  = idx / (HH * DD);
  int rem = idx % (HH * DD);
  int h = rem / DD, c = rem % DD;
  const float* W  = qk ? Wk : Wq;
  const float* wa = Wa + qk * HD;
  float acc = 0.f;
#pragma unroll 8
  for (int d = 0; d < HD; ++d) acc += wa[d] * W[(h * HD + d) * DD + c];
  weff[idx] = acc;
}

// ---------------------------------------------------------------
// Kernel B: lq[b,h,i] = X_row . weff_q[h] + (bq[h,:].wq_a) ; same for lk.
// layout: lout[((qk*32 + b)*12 + h)*128 + s]
__global__ __launch_bounds__(256) void lqk_kernel(const float* __restrict__ X,
                                                  const float* __restrict__ weff,
                                                  const float* __restrict__ bq,
                                                  const float* __restrict__ bk,
                                                  const float* __restrict__ Wa,
                                                  float* __restrict__ lout) {
  int idx = blockIdx.x * 256 + threadIdx.x;         // 0 .. 98303
  int i   = idx / 24;                               // global row 0..4095
  int rem = idx % 24;
  int qk = rem / HH;
  int h  = rem % HH;
  const float* x = X + i * DD;
  const float* w = weff + (qk * HH + h) * DD;
  float acc = 0.f;
#pragma unroll 4
  for (int c = 0; c < DD; c += 4)
    acc += x[c] * w[c] + x[c + 1] * w[c + 1] + x[c + 2] * w[c + 2] + x[c + 3] * w[c + 3];
  const float* bb = qk ? bk : bq;
  const float* wa = Wa + qk * HD;
  float bacc = 0.f;
#pragma unroll 8
  for (int d = 0; d < HD; ++d) bacc += bb[h * HD + d] * wa[d];
  acc += bacc;
  int b = i >> 7, s = i & 127;
  lout[((qk * BB + b) * HH + h) * SEQ + s] = acc;
}

// ---------------------------------------------------------------
// Kernel C: V projection, f32 WMMA.  Y = X(4096x768) @ Wv^T + bv.
// Workgroup = 4 waves, 64x64 output tile. blockIdx.x = head (N tile), blockIdx.y = M tile.
// Output layout: vws[((b*12 + h)*128 + s)*64 + d]
__global__ __launch_bounds__(128) void vproj_kernel(const float* __restrict__ X,
                                                    const float* __restrict__ Wv,
                                                    const float* __restrict__ bv,
                                                    float* __restrict__ vws) {
  int head = blockIdx.x;                 // 0..11  (n0 = head*64)
  int wave = threadIdx.x >> 5;
  int lane = threadIdx.x & 31;
  int m0 = blockIdx.y * 64 + wave * 16;  // row-tile base for this wave
  int lr = lane & 15;
  int kk = (lane >> 4) << 1;             // 0 or 2

  const float* arow = X + (m0 + lr) * DD;            // A[m][k] = X[row][k]
  const float* brow = Wv + (head * HD + lr) * DD;    // B[k][n] = Wv[n_glob][k]

  v8f acc0 = {}, acc1 = {}, acc2 = {}, acc3 = {};
#pragma unroll 2
  for (int k0 = 0; k0 < DD; k0 += 4) {
    v2f a  = *(const v2f*)(arow + k0 + kk);
    v2f b0 = *(const v2f*)(brow + k0 + kk);
    v2f b1 = *(const v2f*)(brow + 16 * DD + k0 + kk);
    v2f b2 = *(const v2f*)(brow + 32 * DD + k0 + kk);
    v2f b3 = *(const v2f*)(brow + 48 * DD + k0 + kk);
    acc0 = WMMA4(a, b0, acc0);
    acc1 = WMMA4(a, b1, acc1);
    acc2 = WMMA4(a, b2, acc2);
    acc3 = WMMA4(a, b3, acc3);
  }
  float bv0 = bv[head * HD + lr];
  float bv1 = bv[head * HD + 16 + lr];
  float bv2 = bv[head * HD + 32 + lr];
  float bv3 = bv[head * HD + 48 + lr];
  int mtop = (lane >> 4) * 8;
#pragma unroll
  for (int r = 0; r < 8; ++r) {
    int i = m0 + mtop + r;
    int b = i >> 7, s = i & 127;
    float* o = vws + (((b * HH + head) * SEQ + s) * HD);
    o[lr]      = acc0[r] + bv0;
    o[16 + lr] = acc1[r] + bv1;
    o[32 + lr] = acc2[r] + bv2;
    o[48 + lr] = acc3[r] + bv3;
  }
}

// ---------------------------------------------------------------
// Kernel D: per (b,h) block: edge scalars + softmax stats + attn out + ctx = attn @ v (WMMA).
__global__ __launch_bounds__(256) void attn_kernel(const float* __restrict__ desc,
                                                   const float* __restrict__ vws,
                                                   const float* __restrict__ lqws,
                                                   const float* __restrict__ lkws,
                                                   const float* __restrict__ Wa,
                                                   const float* __restrict__ ba_p,
                                                   float* __restrict__ out_basis,
                                                   float* __restrict__ out_attn) {
  __shared__ float v_s[SEQ * HD];     // 32 KB
  __shared__ float lq_s[SEQ], lk_s[SEQ];
  __shared__ float eh_s[SEQ], ec_s[SEQ];
  __shared__ float base_s[SEQ], inv_s[SEQ];

  int bh = blockIdx.x;
  int b = bh / HH, h = bh % HH;
  int t = threadIdx.x;

  // stage V tile (128x64) into LDS
  const float* vsrc = vws + (size_t)bh * SEQ * HD;
  for (int i = t; i < SEQ * HD / 4; i += 256) ((f4*)v_s)[i] = ((const f4*)vsrc)[i];

  // lq / lk
  if (t < SEQ) lq_s[t] = lqws[bh * SEQ + t];
  else         lk_s[t - SEQ] = lkws[bh * SEQ + (t - SEQ)];

  // edge scalars: eh_s[s] = desc[b,s,h,:] . we_a ; ec_s uses head h%6 (CLS row)
  const float* wea = Wa + 2 * HD;
  if (t < SEQ) {
    float e = 0.f;
    if (t < SS) {
      const float* dp = desc + ((size_t)b * SS + t) * DD + h * HD;
#pragma unroll 8
      for (int d = 0; d < HD; ++d) e += dp[d] * wea[d];
    }
    eh_s[t] = e;
  } else {
    int s = t - SEQ;
    int hc = h % 6;
    float e = 0.f;
    if (s < SS) {
      const float* dp = desc + ((size_t)b * SS + s) * DD + hc * HD;
#pragma unroll 8
      for (int d = 0; d < HD; ++d) e += dp[d] * wea[d];
    }
    ec_s[s] = e;
  }
  __syncthreads();

  float ba = ba_p[0];

  // softmax: thread r < 128 owns row r; store (base - max) and 1/sum; write attn to global
  if (t < SEQ) {
    int r = t;
    float rowterm = (r >= 1 && h < 6) ? eh_s[r - 1] : 0.f;
    float bse = lq_s[r] + ba + rowterm;
    bool iscls = (r == 0);
    bool cflag = (h >= 6);
    float mx = -3.4e38f;
    for (int j = 1; j < SEQ; ++j) {
      float ct = iscls ? ec_s[j - 1] : (cflag ? eh_s[j - 1] : 0.f);
      mx = fmaxf(mx, bse + lk_s[j] + ct);
    }
    float sum = 0.f;
    for (int j = 1; j < SEQ; ++j) {
      float ct = iscls ? ec_s[j - 1] : (cflag ? eh_s[j - 1] : 0.f);
      sum += __expf(bse + lk_s[j] + ct - mx);
    }
    float inv = 1.f / sum;
    base_s[r] = bse - mx;
    inv_s[r]  = inv;
    float* gout = out_attn + ((size_t)bh * SEQ + r) * SEQ;
    gout[0] = 0.f;                                   // masked column 0
    for (int j = 1; j < SEQ; ++j) {
      float ct = iscls ? ec_s[j - 1] : (cflag ? eh_s[j - 1] : 0.f);
      gout[j] = __expf(bse + lk_s[j] + ct - mx) * inv;
    }
  }
  __syncthreads();

  // ctx = attn(128x128) @ v(128x64) with f32 WMMA; wave w -> rows 16w..16w+15, all 64 cols.
  int wave = t >> 5, lane = t & 31;
  int lr = lane & 15;
  int kk = (lane >> 4) << 1;
  int m = wave * 16 + lr;              // every lane's A-row
  float bse = base_s[m];
  float scl = inv_s[m];
  bool iscls = (m == 0);
  bool cflag = (h >= 6);

  v8f a0 = {}, a1 = {}, a2 = {}, a3 = {};
  for (int k0 = 0; k0 < SEQ; k0 += 4) {
    int k = k0 + kk;
    // recompute attention values (branchless selects keep EXEC full for WMMA)
    int km0 = (k > 0) ? k - 1 : 0;
    float ct0 = iscls ? ec_s[km0] : (cflag ? eh_s[km0] : 0.f);
    float e0 = __expf(bse + lk_s[k] + ct0) * scl;
    float ct1 = iscls ? ec_s[k] : (cflag ? eh_s[k] : 0.f);
    float e1 = __expf(bse + lk_s[k + 1] + ct1) * scl;
    v2f af;
    af.x = (k == 0) ? 0.f : e0;
    af.y = e1;
    const float* vp = v_s + k * HD + lr;
    v2f b0; b0.x = vp[0];  b0.y = vp[HD];
    v2f b1; b1.x = vp[16]; b1.y = vp[HD + 16];
    v2f b2; b2.x = vp[32]; b2.y = vp[HD + 32];
    v2f b3; b3.x = vp[48]; b3.y = vp[HD + 48];
    a0 = WMMA4(af, b0, a0);
    a1 = WMMA4(af, b1, a1);
    a2 = WMMA4(af, b2, a2);
    a3 = WMMA4(af, b3, a3);
  }
  int mtop = (lane >> 4) * 8;
#pragma unroll
  for (int r = 0; r < 8; ++r) {
    int s = wave * 16 + mtop + r;
    float* o = out_basis + (((size_t)b * SEQ + s) * HH + h) * HD;
    o[lr]      = a0[r];
    o[16 + lr] = a1[r];
    o[32 + lr] = a2[r];
    o[48 + lr] = a3[r];
  }
}

// ---------------------------------------------------------------
extern "C" void kernel_launch(void* const* d_in, const int* in_sizes, int n_in,
                              void* d_out, int out_size, void* d_ws, size_t ws_size,
                              hipStream_t stream) {
  const float* desc = (const float*)d_in[0];
  const float* nv   = (const float*)d_in[1];
  const float* Wq   = (const float*)d_in[2];
  const float* bq   = (const float*)d_in[3];
  const float* Wk   = (const float*)d_in[4];
  const float* bk   = (const float*)d_in[5];
  const float* Wv   = (const float*)d_in[6];
  const float* bv   = (const float*)d_in[7];
  const float* Wa   = (const float*)d_in[8];
  const float* ba   = (const float*)d_in[9];

  float* out = (float*)d_out;
  float* ws  = (float*)d_ws;
  float* vws  = ws + VWS_OFF;
  float* weff = ws + WEFF_OFF;
  float* lq   = ws + LQ_OFF;            // lk lives at lq + 49152 (contiguous, indexed by qk)
  float* out_basis = out;               // (B,128,H,64) = 3145728 floats
  float* out_attn  = out + 3145728;     // (B,H,128,128)

  hipLaunchKernelGGL(weff_kernel, dim3(72), dim3(256), 0, stream, Wq, Wk, Wa, weff);
  hipLaunchKernelGGL(lqk_kernel, dim3(384), dim3(256), 0, stream, nv, weff, bq, bk, Wa, lq);
  hipLaunchKernelGGL(vproj_kernel, dim3(12, 64), dim3(128), 0, stream, nv, Wv, bv, vws);
  hipLaunchKernelGGL(attn_kernel, dim3(BB * HH), dim3(256), 0, stream,
                     desc, vws, lq, lq + 49152, Wa, ba, out_basis, out_attn);
}